// HistogramLoss_3444563772224
// MI455X (gfx1250) — compile-verified
//
#include <hip/hip_runtime.h>

// ---------------------------------------------------------------------------
// HistogramLoss for MI455X (gfx1250, wave32).
//
// Roofline: 2 x 64MB streamed once -> memory-bound, ~5.5us at 23.3 TB/s.
// Strategy:
//   K1: zero the 202-entry u32 workspace histogram.
//   K2: grid-stride b128 non-temporal loads, floor(x*100) binning
//       (v_cvt_flr_i32_f32, avoids precise-div VALU chain), per-wave
//       privatized LDS histograms (ds_add_u32), block reduce -> global atomics.
//   K3: single wave32 block. Reversed cumsum of 100 bins done as
//       D = UpperTri(16x16) x ChunkMatrix(16x16) via 4 chained
//       V_WMMA_F32_16X16X4_F32 ops (scan-as-matmul), cross-chunk fixup,
//       then normalization + KL reduction. EXEC all-ones (32-thread block).
// ---------------------------------------------------------------------------

typedef __attribute__((ext_vector_type(2))) float v2f;
typedef __attribute__((ext_vector_type(4))) float v4f;
typedef __attribute__((ext_vector_type(8))) float v8f;

#define HL_BINS 100
#define HL_NB   101   // 100 bins + overflow slot (idx < 0)

// ---------------------------------------------------------------- K1: zero ws
__global__ __launch_bounds__(256) void hl_zero(unsigned int* __restrict__ g, int n) {
  int i = blockIdx.x * blockDim.x + threadIdx.x;
  if (i < n) g[i] = 0u;
}

// ------------------------------------------------------------- K2: histogram
__device__ __forceinline__ void hl_bin(unsigned int* h, float x) {
  // reference: idx = floor((x - MIN)/DELTA), MIN=0, DELTA=0.01
  int idx = __float2int_rd(x * 100.0f);          // v_cvt_flr_i32_f32
  int id  = (idx < 0) ? HL_BINS : (idx > HL_BINS - 1 ? HL_BINS - 1 : idx);
  atomicAdd(&h[id], 1u);                          // ds_add_u32 (no rtn)
}

__global__ __launch_bounds__(256) void hl_hist(const v4f* __restrict__ pred,
                                               const v4f* __restrict__ targ,
                                               unsigned int* __restrict__ ghist,
                                               int n4) {
  // 8 waves x {pred,targ} x 104 bins (stride 104 staggers 64 LDS banks)
  __shared__ unsigned int lh[8][2][104];
  unsigned int* flat = &lh[0][0][0];
  for (int i = threadIdx.x; i < 8 * 2 * 104; i += 256) flat[i] = 0u;
  __syncthreads();

  const int wave = threadIdx.x >> 5;              // wave32
  unsigned int* hp = &lh[wave][0][0];
  unsigned int* ht = &lh[wave][1][0];

  const int stride = gridDim.x * blockDim.x;
  for (int i = blockIdx.x * blockDim.x + threadIdx.x; i < n4; i += stride) {
    v4f v = __builtin_nontemporal_load(&pred[i]); // global_load_b128, TH=NT
    hl_bin(hp, v.x); hl_bin(hp, v.y); hl_bin(hp, v.z); hl_bin(hp, v.w);
    v4f w = __builtin_nontemporal_load(&targ[i]);
    hl_bin(ht, w.x); hl_bin(ht, w.y); hl_bin(ht, w.z); hl_bin(ht, w.w);
  }
  __syncthreads();

  for (int b = threadIdx.x; b < HL_NB; b += 256) {
    unsigned int sp = 0, st = 0;
    for (int w = 0; w < 8; ++w) { sp += lh[w][0][b]; st += lh[w][1][b]; }
    if (sp) atomicAdd(&ghist[b], sp);             // global_atomic_add_u32
    if (st) atomicAdd(&ghist[HL_NB + b], st);
  }
}

// ------------------------------------------- K3: WMMA suffix-scan + KL (1 wave)
__global__ __launch_bounds__(32) void hl_finalize(const unsigned int* __restrict__ ghist,
                                                  float* __restrict__ out) {
  __shared__ float Mp[112], Mt[112];   // 7 chunks of 16 (bins 0..99, zero-pad)
  __shared__ float Sp[16], St[16];     // per-chunk totals S[c] = D[0][c]
  __shared__ float redA[32], redB[32];

  const int lane = threadIdx.x;
  for (int i = lane; i < 112; i += 32) {
    Mp[i] = (i < HL_BINS) ? (float)ghist[i] : 0.0f;
    Mt[i] = (i < HL_BINS) ? (float)ghist[HL_NB + i] : 0.0f;
  }
  __syncthreads();

  const int n    = lane & 15;   // A: row m; B/C/D: column n
  const int half = lane >> 4;   // lane group 0..15 vs 16..31

  // D = L x M, L[i][k] = (k >= i): within-chunk reversed cumsum of each column.
  // K=16 split into 4 K-blocks of 4, accumulated through C.
  v8f dp = {}; v8f dt = {};
  for (int kb = 0; kb < 4; ++kb) {
    const int k0 = 4 * kb + 2 * half;     // A layout: lanes16-31 hold K=+2,+3
    v2f a, bp, bt;
    a.x = (k0     >= n) ? 1.0f : 0.0f;    // L[m][k0],   m = lane&15
    a.y = (k0 + 1 >= n) ? 1.0f : 0.0f;    // L[m][k0+1]
    // B layout (row-striped): VGPR0 = row k0, VGPR1 = row k0+1 for this half.
    bp.x = Mp[n * 16 + k0];  bp.y = Mp[n * 16 + k0 + 1];
    bt.x = Mt[n * 16 + k0];  bt.y = Mt[n * 16 + k0 + 1];
    dp = __builtin_amdgcn_wmma_f32_16x16x4_f32(false, a, false, bp, (short)0, dp, false, false);
    dt = __builtin_amdgcn_wmma_f32_16x16x4_f32(false, a, false, bt, (short)0, dt, false, false);
  }

  // dp[r] = D[r + 8*half][n]; S[c] = D[0][c] = total of chunk c.
  if (half == 0) { Sp[n] = dp[0]; St[n] = dt[0]; }
  __syncthreads();

  float sufp = 0.0f, suft = 0.0f;         // sum of chunk totals after chunk n
  for (int c = n + 1; c < 7; ++c) { sufp += Sp[c]; suft += St[c]; }

  // cum[g] = D[i][c] + suffix(c), g = c*16 + i, valid for g < 100, c < 7.
  float psum = 0.0f, tsum = 0.0f;
  float cp[8], ct[8];
  for (int r = 0; r < 8; ++r) {
    const int g = n * 16 + r + 8 * half;
    const bool valid = (n < 7) && (g < HL_BINS);
    cp[r] = valid ? (dp[r] + sufp) : 0.0f;
    ct[r] = valid ? (dt[r] + suft) : 0.0f;
    psum += cp[r]; tsum += ct[r];
  }
  redA[lane] = psum; redB[lane] = tsum;
  __syncthreads();
  if (lane == 0) {
    float a = 0.0f, b = 0.0f;
    for (int i = 0; i < 32; ++i) { a += redA[i]; b += redB[i]; }
    redA[0] = a; redB[0] = b;
  }
  __syncthreads();
  psum = redA[0]; tsum = redB[0];

  // kl = sum( xlogy(t,t) - t*log(p) ) / BINS,  p = cum_p/psum, t = cum_t/tsum
  float kl = 0.0f;
  for (int r = 0; r < 8; ++r) {
    const int g = n * 16 + r + 8 * half;
    if ((n < 7) && (g < HL_BINS)) {
      const float p = cp[r] / psum;
      const float t = ct[r] / tsum;
      const float xlogy = (t > 0.0f) ? t * logf(t) : 0.0f;
      kl += xlogy - t * logf(p);
    }
  }
  __syncthreads();
  redA[lane] = kl;
  __syncthreads();
  if (lane == 0) {
    float a = 0.0f;
    for (int i = 0; i < 32; ++i) a += redA[i];
    out[0] = a / (float)HL_BINS;
  }
}

// ---------------------------------------------------------------------------
extern "C" void kernel_launch(void* const* d_in, const int* in_sizes, int n_in,
                              void* d_out, int out_size, void* d_ws, size_t ws_size,
                              hipStream_t stream) {
  const float* pred = (const float*)d_in[0];
  const float* targ = (const float*)d_in[1];
  const int n  = in_sizes[0];          // 64*1*512*512 = 2^24, divisible by 4
  const int n4 = n >> 2;

  unsigned int* ghist = (unsigned int*)d_ws;  // [0..100]=pred, [101..201]=targ

  hl_zero<<<1, 256, 0, stream>>>(ghist, 2 * HL_NB);

  // 4096 blocks x 8 waves = 32K waves: enough to saturate HBM; ~4 float4
  // iterations per thread per input.
  const int blocks = 4096;
  hl_hist<<<blocks, 256, 0, stream>>>((const v4f*)pred, (const v4f*)targ, ghist, n4);

  hl_finalize<<<1, 32, 0, stream>>>(ghist, (float*)d_out);
}